// Diagomal_DWConv_62130996904232
// MI455X (gfx1250) — compile-verified
//
#include <hip/hip_runtime.h>
#include <hip/hip_bf16.h>
#include <stdint.h>

// Problem constants (fixed by the reference): x is [8, 64, 256, 256] fp32.
#define KS    5
#define WCOL  256                 // columns (W)
#define RTOT  (8 * 64 * 256)     // flattened rows B*C*H = 131072
#define TILE  32                  // output rows per block (divides 256 -> one channel per tile)
#define HALO  2
#define LROWS (TILE + 2 * HALO)   // 36 rows staged in LDS (36 KB)

typedef unsigned int v4u __attribute__((ext_vector_type(4)));
typedef int          v8i __attribute__((ext_vector_type(8)));
typedef int          v4i __attribute__((ext_vector_type(4)));

__global__ __launch_bounds__(256) void diag_dwconv_tdm_kernel(
    const float* __restrict__ x, const float* __restrict__ weight,
    const float* __restrict__ bias, float* __restrict__ out)
{
    __shared__ float tile[LROWS * WCOL];   // 36 KB -> 8 blocks/WGP resident

    const int tid    = threadIdx.x;        // 256 threads = 8 waves (wave32)
    const int r0     = blockIdx.x * TILE;  // first output row of this block
    const int gstart = r0 - HALO;          // first x row needed (may be <0)

    // ---- TDM first: bulk 2-D tile DMA global -> LDS (clipped to valid rows).
    // Issued by wave 0 only (tensor ops ignore EXEC; one wave = one DMA).
    // Waiting is deferred so the DMA overlaps the halo patch + scalar loads.
    if (tid < 32) {
        int ls = gstart < 0 ? 0 : gstart;
        int le = gstart + LROWS; if (le > RTOT) le = RTOT;
        const int nrows = le - ls;          // rows the TDM moves
        const int off   = ls - gstart;      // LDS row offset of first moved row

        const uint64_t ga  = (uint64_t)(uintptr_t)x + (uint64_t)ls * (WCOL * 4);
        const uint32_t lds = (uint32_t)(uintptr_t)&tile[0] + (uint32_t)off * (WCOL * 4);

        // D# group 0 (ISA 8.3): count=1 | lds_addr | global_addr[56:0] | type=2
        v4u g0;
        g0[0] = 1u;                                   // count=1, user descriptor
        g0[1] = lds;                                  // LDS byte address
        g0[2] = (uint32_t)ga;                         // global_addr[31:0]
        g0[3] = (uint32_t)((ga >> 32) & 0x01FFFFFFull) | 0x80000000u; // [56:32] | type=2

        // D# group 1 (ISA 8.4)
        v8i g1;
        g1[0] = (int)(2u << 16);                                  // data_size=4B; wg_mask=0
        g1[1] = (int)(((uint32_t)WCOL & 0xFFFFu) << 16);          // tensor_dim0[15:0]
        g1[2] = (int)(((uint32_t)WCOL >> 16) |
                      (((uint32_t)RTOT & 0xFFFFu) << 16));        // dim0 hi | tensor_dim1 lo
        g1[3] = (int)(((uint32_t)RTOT >> 16) |
                      ((uint32_t)WCOL << 16));                    // tensor_dim1 hi | tile_dim0
        g1[4] = (int)((uint32_t)nrows & 0xFFFFu);                 // tile_dim1; tile_dim2=0
        g1[5] = (int)WCOL;                                        // tensor_dim0_stride
        g1[6] = 0;                                                // stride hi | dim1_stride lo
        g1[7] = 0;                                                // dim1_stride hi

        v4i g2 = {0, 0, 0, 0};   // 2-D tensor: groups 2/3 unused
        v4i g3 = {0, 0, 0, 0};
#if defined(__clang_major__) && (__clang_major__ >= 23)
        v8i gx = {0, 0, 0, 0, 0, 0, 0, 0};
        __builtin_amdgcn_tensor_load_to_lds(g0, g1, g2, g3, gx, 0);
#else
        __builtin_amdgcn_tensor_load_to_lds(g0, g1, g2, g3, 0);
#endif
    }

    // ---- wraparound halo patch (only first / last block ever takes these);
    // rows written here are exactly the rows the TDM was clipped away from,
    // so the two writers touch disjoint LDS rows.
    if (gstart < 0) {
        for (int j = 0; j < -gstart; ++j)
            tile[j * WCOL + tid] = x[(size_t)(RTOT + gstart + j) * WCOL + tid];
    }
    {
        const int over = gstart + LROWS - RTOT;
        if (over > 0) {
            for (int j = LROWS - over; j < LROWS; ++j)
                tile[j * WCOL + tid] = x[(size_t)(gstart + j - RTOT) * WCOL + tid];
        }
    }

    // ---- per-block channel: 32-row aligned tiles never cross a channel ----
    const int   c  = (r0 >> 8) & 63;
    const float w0 = weight[c * KS + 0];
    const float w1 = weight[c * KS + 1];
    const float w2 = weight[c * KS + 2];
    const float w3 = weight[c * KS + 3];
    const float w4 = weight[c * KS + 4];
    const float bz = bias[c];

    // Complete the DMA, then publish LDS to all 8 waves.
    if (tid < 32) {
        __builtin_amdgcn_s_wait_tensorcnt(0);
    }
    __syncthreads();

    // one column per lane -> stride-1, bank-conflict-free LDS reads
    const int col = tid;
    float* op = out + (size_t)r0 * WCOL + col;

    #pragma unroll 4
    for (int i = 0; i < TILE; ++i) {
        // out[r0+i][col] = bz + sum_k w_k * X[r0+i-(k-2)][col+(k-2)]
        // LDS row for tap k is (i + 4 - k); column is col + k - 2 (zero-padded).
        float acc = bz;
        {   // k=0: row i+4, col-2
            const int cc = col - 2;
            const float v = (cc >= 0) ? tile[(i + 4) * WCOL + cc] : 0.0f;
            acc = fmaf(w0, v, acc);
        }
        {   // k=1: row i+3, col-1
            const int cc = col - 1;
            const float v = (cc >= 0) ? tile[(i + 3) * WCOL + cc] : 0.0f;
            acc = fmaf(w1, v, acc);
        }
        // k=2: row i+2, col
        acc = fmaf(w2, tile[(i + 2) * WCOL + col], acc);
        {   // k=3: row i+1, col+1
            const int cc = col + 1;
            const float v = (cc < WCOL) ? tile[(i + 1) * WCOL + cc] : 0.0f;
            acc = fmaf(w3, v, acc);
        }
        {   // k=4: row i, col+2
            const int cc = col + 2;
            const float v = (cc < WCOL) ? tile[i * WCOL + cc] : 0.0f;
            acc = fmaf(w4, v, acc);
        }
        // NT store: stream the 134 MB output past L2 so x (134 MB) stays
        // L2-resident (192 MB) for halo / adjacent-tile re-reads.
        __builtin_nontemporal_store(acc, op + (size_t)i * WCOL);
    }
}

extern "C" void kernel_launch(void* const* d_in, const int* in_sizes, int n_in,
                              void* d_out, int out_size, void* d_ws, size_t ws_size,
                              hipStream_t stream) {
    const float* x      = (const float*)d_in[0];  // [8,64,256,256]
    const float* weight = (const float*)d_in[1];  // [64,1,5]
    const float* bias   = (const float*)d_in[2];  // [64]
    float*       out    = (float*)d_out;          // [8,64,256,256]

    dim3 grid(RTOT / TILE);   // 4096 blocks
    dim3 block(256);          // 8 wave32 waves
    diag_dwconv_tdm_kernel<<<grid, block, 0, stream>>>(x, weight, bias, out);
    (void)in_sizes; (void)n_in; (void)out_size; (void)d_ws; (void)ws_size;
}